// DynamicScatterCustom_6030134083613
// MI455X (gfx1250) — compile-verified
//
#include <hip/hip_runtime.h>
#include <cstdint>

#define NXc 352
#define NYc 400
#define NZc 20
#define NBc 4
#define GRIDC (NBc * NZc * NYc * NXc)   // 11,264,000 cells
#define PPW 8                            // point-pairs per wave in scatter pass

typedef float    v4f __attribute__((ext_vector_type(4)));
typedef uint32_t v4u __attribute__((ext_vector_type(4)));

// ---------------------------------------------------------------------------
// 1) init output: voxels region = 0.0f, coors region = -1.0f  (NT b128 stores)
// ---------------------------------------------------------------------------
__global__ void k_init_out(v4f* __restrict__ out4, int n) {
  long long i   = (long long)blockIdx.x * 256 + threadIdx.x;
  long long nv  = (long long)n * 16;         // voxels: n*64 floats = n*16 v4f
  long long tot = (long long)n * 17;         // + coors: n*4 floats = n v4f
  if (i >= tot) return;
  float f = (i < nv) ? 0.f : -1.f;
  v4f v = {f, f, f, f};
  __builtin_nontemporal_store(v, &out4[i]);  // write-once: don't pollute L2
}

// ---------------------------------------------------------------------------
// 2) zero dense counts grid (GRID/4 v4u writes, exact coverage; stays in L2)
// ---------------------------------------------------------------------------
__global__ void k_zero_counts(v4u* __restrict__ c4) {
  int i = blockIdx.x * 256 + threadIdx.x;
  v4u z = {0u, 0u, 0u, 0u};
  c4[i] = z;                                 // RT: grid must stay L2-resident
}

// ---------------------------------------------------------------------------
// 3) per-point histogram into dense grid
// ---------------------------------------------------------------------------
__global__ void k_count(const int4* __restrict__ coors,
                        uint32_t* __restrict__ counts, int n) {
  int i = blockIdx.x * 256 + threadIdx.x;
  if (i >= n) return;
  int4 c = coors[i];   // {b, z, y, x}
  int key = ((c.x * NZc + c.y) * NYc + c.z) * NXc + c.w;
  atomicAdd(&counts[key], 1u);
}

// ---------------------------------------------------------------------------
// 4) scan pass 1: per-block occupancy totals (1024 cells / block)
// ---------------------------------------------------------------------------
__global__ void k_scan1(const uint4* __restrict__ counts4,
                        uint32_t* __restrict__ blockTotals) {
  __shared__ uint32_t sh[256];
  int gid = blockIdx.x * 256 + threadIdx.x;
  uint4 c = counts4[gid];
  uint32_t s = (c.x > 0u) + (c.y > 0u) + (c.z > 0u) + (c.w > 0u);
  sh[threadIdx.x] = s;
  __syncthreads();
  for (int off = 128; off > 0; off >>= 1) {
    if (threadIdx.x < off) sh[threadIdx.x] += sh[threadIdx.x + off];
    __syncthreads();
  }
  if (threadIdx.x == 0) blockTotals[blockIdx.x] = sh[0];
}

// ---------------------------------------------------------------------------
// 5) scan pass 2: exclusive scan of 11,000 block totals (single block, 1024T)
// ---------------------------------------------------------------------------
__global__ void k_scan2(uint32_t* __restrict__ data, int nblocks) {
  __shared__ uint32_t sh[1024];
  __shared__ uint32_t carry_s;
  int t = threadIdx.x;
  if (t == 0) carry_s = 0u;
  __syncthreads();
  for (int base = 0; base < nblocks; base += 1024) {
    int i = base + t;
    uint32_t v = (i < nblocks) ? data[i] : 0u;
    sh[t] = v;
    __syncthreads();
    for (int off = 1; off < 1024; off <<= 1) {       // inclusive Hillis-Steele
      uint32_t x = (t >= off) ? sh[t - off] : 0u;
      __syncthreads();
      sh[t] += x;
      __syncthreads();
    }
    uint32_t carry = carry_s;
    if (i < nblocks) data[i] = carry + sh[t] - v;    // exclusive + carry
    __syncthreads();
    if (t == 0) carry_s = carry + sh[1023];
    __syncthreads();
  }
}

// ---------------------------------------------------------------------------
// 6) scan pass 3: final ranks + decoded voxel_coors rows
// ---------------------------------------------------------------------------
__device__ __forceinline__ void emit_voxel(uint32_t cell, uint32_t r,
                                           uint32_t* __restrict__ ranks,
                                           v4f* __restrict__ outCoors4) {
  ranks[cell] = r;                                   // RT: L2-resident
  uint32_t x = cell % NXc;
  uint32_t t = cell / NXc;
  uint32_t y = t % NYc;
  t /= NYc;
  uint32_t z = t % NZc;
  uint32_t b = t / NZc;
  v4f v = {(float)b, (float)z, (float)y, (float)x};
  __builtin_nontemporal_store(v, &outCoors4[r]);     // write-once output
}

__global__ void k_scan3(const uint4* __restrict__ counts4,
                        const uint32_t* __restrict__ blockOffs,
                        uint32_t* __restrict__ ranks,
                        v4f* __restrict__ outCoors4) {
  __shared__ uint32_t sh[256];
  int t = threadIdx.x;
  int gid = blockIdx.x * 256 + t;
  uint4 c = counts4[gid];
  uint32_t f0 = c.x > 0u, f1 = c.y > 0u, f2 = c.z > 0u, f3 = c.w > 0u;
  uint32_t ts = f0 + f1 + f2 + f3;
  sh[t] = ts;
  __syncthreads();
  for (int off = 1; off < 256; off <<= 1) {
    uint32_t v = (t >= off) ? sh[t - off] : 0u;
    __syncthreads();
    sh[t] += v;
    __syncthreads();
  }
  uint32_t r = sh[t] - ts + blockOffs[blockIdx.x];   // exclusive rank base
  uint32_t cell = (uint32_t)gid * 4u;
  if (f0) { emit_voxel(cell + 0u, r, ranks, outCoors4); r++; }
  if (f1) { emit_voxel(cell + 1u, r, ranks, outCoors4); r++; }
  if (f2) { emit_voxel(cell + 2u, r, ranks, outCoors4); r++; }
  if (f3) { emit_voxel(cell + 3u, r, ranks, outCoors4); }
}

// ---------------------------------------------------------------------------
// 7) scatter pass: async global->LDS double-buffered feature streaming
// ---------------------------------------------------------------------------
__device__ __forceinline__ void accum_point(v4f v, long long pt, int lane15,
                                            const int4* __restrict__ coors,
                                            const uint32_t* __restrict__ counts,
                                            const uint32_t* __restrict__ ranks,
                                            float* __restrict__ out) {
  int4 c = coors[pt];                                  // same-addr lanes coalesce
  int key = ((c.x * NZc + c.y) * NYc + c.z) * NXc + c.w;
  uint32_t cnt = counts[key];                          // L2-resident gather
  uint32_t r   = ranks[key];
  float* dst = out + (size_t)r * 64 + (size_t)lane15 * 4;
  if (cnt == 1u) {
    // ~91% of points: sole occupant -> plain NT b128 store (scale == 1)
    __builtin_nontemporal_store(v, reinterpret_cast<v4f*>(dst));
  } else {
    float s = 1.0f / (float)cnt;
    atomicAdd(dst + 0, v.x * s);
    atomicAdd(dst + 1, v.y * s);
    atomicAdd(dst + 2, v.z * s);
    atomicAdd(dst + 3, v.w * s);
  }
}

__global__ __launch_bounds__(256) void k_scatter(
    const float* __restrict__ points, const int4* __restrict__ coors,
    const uint32_t* __restrict__ counts, const uint32_t* __restrict__ ranks,
    float* __restrict__ out, int n, int totalPairs) {
  __shared__ v4f stage[8][2][32];                      // 8 waves x 2 bufs x 512B
  const int lane = threadIdx.x & 31;
  const int w    = threadIdx.x >> 5;
  const long long waveId = (long long)blockIdx.x * 8 + w;
  const long long pair0  = waveId * PPW;
  if (pair0 >= totalPairs) return;                     // wave-uniform exit

  // LDS byte offsets for this lane's 16B slot in each buffer (generic shared
  // pointer low 32 bits == group-relative LDS offset on gfx1250).
  uint32_t lds0 = (uint32_t)(uintptr_t)&stage[w][0][lane];
  uint32_t lds1 = (uint32_t)(uintptr_t)&stage[w][1][lane];

  auto issue = [&](long long p, int buf) {
    // lane l DMAs bytes [p*512 + l*16, +16) -> contiguous 512B = 2 point rows.
    // th:TH_LOAD_NT: features are consumed exactly once; keep grids in L2.
    const float* g = points + (size_t)p * 128 + (size_t)lane * 4;
    uint32_t lo = buf ? lds1 : lds0;
    asm volatile("global_load_async_to_lds_b128 %0, %1, off th:TH_LOAD_NT"
                 :: "v"(lo), "v"(g) : "memory");       // ASYNCcnt++
  };

  auto full = [&](long long p) {
    return (p < (long long)totalPairs) && (2 * p + 2 <= (long long)n);
  };

  bool have_cur = full(pair0);
  if (have_cur) issue(pair0, 0);
  int buf = 0;
  for (int i = 0; i < PPW; ++i) {
    long long p = pair0 + i;
    if (p >= (long long)totalPairs) break;
    bool have_next = (i + 1 < PPW) && full(p + 1);
    if (have_next) issue(p + 1, buf ^ 1);              // prefetch next pair
    if (have_cur) {
      if (have_next) asm volatile("s_wait_asynccnt 1" ::: "memory");
      else           asm volatile("s_wait_asynccnt 0" ::: "memory");
      v4f v = stage[w][buf][lane];                     // ds_load_b128
      long long pt = 2 * p + (lane >> 4);
      accum_point(v, pt, lane & 15, coors, counts, ranks, out);
    } else {
      // odd-n tail: partial pair, direct NT global loads (no asm in divergence)
      long long pt = 2 * p + (lane >> 4);
      if (pt < (long long)n) {
        const v4f* src = reinterpret_cast<const v4f*>(points) + pt * 16 + (lane & 15);
        v4f v = __builtin_nontemporal_load(src);
        accum_point(v, pt, lane & 15, coors, counts, ranks, out);
      }
    }
    have_cur = have_next;
    buf ^= 1;
  }
}

// ---------------------------------------------------------------------------
extern "C" void kernel_launch(void* const* d_in, const int* in_sizes, int n_in,
                              void* d_out, int out_size, void* d_ws, size_t ws_size,
                              hipStream_t stream) {
  const float* points = (const float*)d_in[0];
  const int4*  coors  = (const int4*)d_in[1];
  float* out = (float*)d_out;
  const int n = in_sizes[0] / 64;

  uint8_t* ws = (uint8_t*)d_ws;
  uint32_t* counts = (uint32_t*)ws;                          // GRID u32 (45 MB)
  uint32_t* ranks  = (uint32_t*)(ws + (size_t)GRIDC * 4);    // GRID u32 (45 MB)
  uint32_t* btot   = (uint32_t*)(ws + (size_t)GRIDC * 8);    // 11,000 u32

  const int NBLK = GRIDC / 1024;                             // 11,000 exact

  {
    long long tot = (long long)n * 17;
    k_init_out<<<(int)((tot + 255) / 256), 256, 0, stream>>>((v4f*)out, n);
  }
  k_zero_counts<<<GRIDC / 4 / 256, 256, 0, stream>>>((v4u*)counts);
  k_count<<<(n + 255) / 256, 256, 0, stream>>>(coors, counts, n);
  k_scan1<<<NBLK, 256, 0, stream>>>((const uint4*)counts, btot);
  k_scan2<<<1, 1024, 0, stream>>>(btot, NBLK);
  k_scan3<<<NBLK, 256, 0, stream>>>((const uint4*)counts, btot, ranks,
                                    (v4f*)(out + (size_t)n * 64));

  const int totalPairs = (n + 1) / 2;
  long long waves = ((long long)totalPairs + PPW - 1) / PPW;
  int blocks = (int)((waves + 7) / 8);
  k_scatter<<<blocks, 256, 0, stream>>>(points, coors, counts, ranks, out, n,
                                        totalPairs);
}